// DampedEMA_30322469110540
// MI455X (gfx1250) — compile-verified
//
#include <hip/hip_runtime.h>

// Problem constants (match reference)
#define BB 4
#define TT 4096
#define DD 1024
#define TC 128            // timesteps per chunk
#define NC (TT / TC)      // 32 chunks

typedef float v2f __attribute__((ext_vector_type(2)));
typedef float v8f __attribute__((ext_vector_type(8)));

// exact-ish beta^e for small integer e (e in [0,16]), branch-free
__device__ __forceinline__ float ipow_beta(float beta, int e) {
    float r = 1.f, p = beta;
    #pragma unroll
    for (int bit = 0; bit < 5; ++bit) {
        r = (e & (1 << bit)) ? r * p : r;
        p *= p;
    }
    return r;
}

// ---------------------------------------------------------------------------
// Phase 1: per-(b,chunk,d) local final state: s = alpha*x + beta*s over TC steps
// ---------------------------------------------------------------------------
__global__ void __launch_bounds__(256)
ema_chunk_reduce(const float* __restrict__ x,
                 const float* __restrict__ alpha_p,
                 float* __restrict__ local_end) {
    const float alpha = alpha_p[0];
    const float beta  = 1.f - alpha;
    const int idx = blockIdx.x * blockDim.x + threadIdx.x;   // b*NC*DD + c*DD + d
    const int d = idx % DD;
    const int c = (idx / DD) % NC;
    const int b = idx / (DD * NC);
    const float* xp = x + ((size_t)(b * TT + c * TC)) * DD + d;
    float s = 0.f;
    #pragma unroll 8
    for (int t = 0; t < TC; ++t) {
        s = fmaf(beta, s, alpha * xp[(size_t)t * DD]);
    }
    local_end[idx] = s;
}

// ---------------------------------------------------------------------------
// Phase 2: per-(b,d) scan across chunk carries; carry_in[c] = true state
//          entering chunk c.  q = beta^TC (TC = 2^7 -> 7 squarings)
// ---------------------------------------------------------------------------
__global__ void __launch_bounds__(256)
ema_carry_scan(const float* __restrict__ local_end,
               const float* __restrict__ alpha_p,
               float* __restrict__ carry_in) {
    const float alpha = alpha_p[0];
    float q = 1.f - alpha;
    #pragma unroll
    for (int i = 0; i < 7; ++i) q *= q;                      // beta^128
    const int idx = blockIdx.x * blockDim.x + threadIdx.x;   // b*DD + d
    const int d = idx % DD;
    const int b = idx / DD;
    float s = 0.f;
    for (int c = 0; c < NC; ++c) {
        const size_t o = ((size_t)b * NC + c) * DD + d;
        carry_in[o] = s;
        s = local_end[o] + q * s;
    }
}

// ---------------------------------------------------------------------------
// Phase 3: main scan.  One wave owns (b, chunk, 16-wide d tile).  Each 16-step
// time tile is H(16x16) = L(16x16) * X(16x16) via 4x V_WMMA_F32_16X16X4_F32,
// plus the carry rank-1 term beta^(i+1) * s_in.
// ---------------------------------------------------------------------------
__global__ void __launch_bounds__(256)
ema_scan_wmma(const float* __restrict__ x,
              const float* __restrict__ carry_in,
              const float* __restrict__ alpha_p,
              float* __restrict__ h) {
    const float alpha = alpha_p[0];
    const float beta  = 1.f - alpha;

    const int lane  = threadIdx.x & 31;
    const int wid   = threadIdx.x >> 5;
    const int gw    = blockIdx.x * 8 + wid;      // global wave id
    const int DT16  = DD / 16;                   // 64 d-tiles
    const int d0    = (gw % DT16) * 16;
    const int c     = (gw / DT16) % NC;
    const int b     = gw / (DT16 * NC);

    const int i     = lane & 15;                 // A row / B,C column within tile
    const int khalf = lane >> 4;                 // lane-half selector

    // A fragments: L[i][j] = (j<=i) ? alpha*beta^(i-j) : 0
    // ISA layout A(16x4 f32): (lane,v) -> M = lane%16, K = 2*(lane/16)+v
    v2f afrag[4];
    #pragma unroll
    for (int kk = 0; kk < 4; ++kk) {
        #pragma unroll
        for (int v = 0; v < 2; ++v) {
            const int j = 4 * kk + 2 * khalf + v;
            afrag[kk][v] = (j <= i) ? alpha * ipow_beta(beta, i - j) : 0.f;
        }
    }

    // carry weights per C register: beta^(M+1), M = v + 8*khalf
    float cw[8];
    #pragma unroll
    for (int v = 0; v < 8; ++v) cw[v] = ipow_beta(beta, v + 8 * khalf + 1);

    // incoming true state for column n = i of this chunk
    float sN = carry_in[((size_t)b * NC + c) * DD + d0 + i];

    const size_t baseT = (size_t)b * TT + (size_t)c * TC;

    for (int tt = 0; tt < TC / 16; ++tt) {
        const int t0 = tt * 16;

        // B fragments: X[t0+K][d0+N]; (lane,v) -> N = lane%16, K = 2*(lane/16)+v
        v2f bfrag[4];
        #pragma unroll
        for (int kk = 0; kk < 4; ++kk) {
            const size_t rowbase = (baseT + (size_t)(t0 + 4 * kk + 2 * khalf)) * DD + d0 + i;
            bfrag[kk][0] = x[rowbase];
            bfrag[kk][1] = x[rowbase + DD];
        }

        v8f acc = {0.f, 0.f, 0.f, 0.f, 0.f, 0.f, 0.f, 0.f};
        #pragma unroll
        for (int kk = 0; kk < 4; ++kk) {
            acc = __builtin_amdgcn_wmma_f32_16x16x4_f32(
                /*neg_a=*/false, afrag[kk],
                /*neg_b=*/false, bfrag[kk],
                /*c_mod=*/(short)0, acc,
                /*reuse_a=*/false, /*reuse_b=*/false);
        }

        // rank-1 carry term: h[M][n] += beta^(M+1) * s_in[n]
        #pragma unroll
        for (int v = 0; v < 8; ++v) acc[v] = fmaf(cw[v], sN, acc[v]);

        // store: C/D layout (lane,v) -> row M = v + 8*khalf, col N = i
        #pragma unroll
        for (int v = 0; v < 8; ++v) {
            h[(baseT + (size_t)(t0 + v + 8 * khalf)) * DD + d0 + i] = acc[v];
        }

        // next tile's carry = row M=15 value (VGPR 7, lanes 16..31)
        sN = __shfl(acc[7], 16 + i, 32);
    }
}

// ---------------------------------------------------------------------------
extern "C" void kernel_launch(void* const* d_in, const int* in_sizes, int n_in,
                              void* d_out, int out_size, void* d_ws, size_t ws_size,
                              hipStream_t stream) {
    (void)in_sizes; (void)n_in; (void)out_size; (void)ws_size;
    const float* x       = (const float*)d_in[0];
    const float* alpha_p = (const float*)d_in[1];
    float* out = (float*)d_out;

    float* local_end = (float*)d_ws;                          // B*NC*DD floats (512 KB)
    float* carry_in  = local_end + (size_t)BB * NC * DD;      // B*NC*DD floats (512 KB)

    // Phase 1: 131072 threads
    ema_chunk_reduce<<<(BB * NC * DD) / 256, 256, 0, stream>>>(x, alpha_p, local_end);
    // Phase 2: 4096 threads
    ema_carry_scan<<<(BB * DD) / 256, 256, 0, stream>>>(local_end, alpha_p, carry_in);
    // Phase 3: 8192 waves, 8 waves/block
    ema_scan_wmma<<<(BB * NC * (DD / 16)) / 8, 256, 0, stream>>>(x, carry_in, alpha_p, out);
}